// TopKLoss_25082609009303
// MI455X (gfx1250) — compile-verified
//
#include <hip/hip_runtime.h>
#include <hip/hip_bf16.h>
#include <float.h>

#define WAVES   8
#define THREADS 256
#define WCHUNK  1024                // floats staged per wave per step (4 KB)
#define LANEF   (WCHUNK / 32)       // 32 floats per lane per step (8 x b128)
#define CHUNK   (WAVES * WCHUNK)    // 8192 floats per block step (32 KB)

#if defined(__has_builtin)
#if __has_builtin(__builtin_amdgcn_global_load_async_to_lds_b128)
#define HAVE_ASYNC_B128 1
#endif
#if __has_builtin(__builtin_amdgcn_s_wait_asynccnt)
#define HAVE_WAIT_ASYNC 1
#endif
#endif

typedef int v4i __attribute__((__vector_size__(16)));
typedef __attribute__((address_space(1))) v4i* gbl_v4i_p;
typedef __attribute__((address_space(3))) v4i* lds_v4i_p;

__device__ __forceinline__ void async_ld16(const float* g, float* l) {
#if defined(HAVE_ASYNC_B128)
  __builtin_amdgcn_global_load_async_to_lds_b128((gbl_v4i_p)g, (lds_v4i_p)l, 0, 0);
#else
  unsigned loff = (unsigned)(unsigned long long)(lds_v4i_p)l;
  asm volatile("global_load_async_to_lds_b128 %0, %1, off"
               :: "v"(loff), "v"(g) : "memory");
#endif
}

__device__ __forceinline__ void wait_async8() {
#if defined(HAVE_WAIT_ASYNC)
  __builtin_amdgcn_s_wait_asynccnt(8);
#else
  asm volatile("s_wait_asynccnt 8" ::: "memory");
#endif
}

__device__ __forceinline__ void wait_async0() {
#if defined(HAVE_WAIT_ASYNC)
  __builtin_amdgcn_s_wait_asynccnt(0);
#else
  asm volatile("s_wait_asynccnt 0" ::: "memory");
#endif
}

// merge two online (max, sumexp) states
__device__ __forceinline__ void sm_merge(float& m, float& s, float m2, float s2) {
  float M = fmaxf(m, m2);
  s = s * __expf(m - M) + s2 * __expf(m2 - M);
  m = M;
}

__global__ __launch_bounds__(THREADS)
void topk_ce_row_kernel(const float* __restrict__ logits,
                        const long long* __restrict__ target,
                        const int* __restrict__ kptr,
                        float* __restrict__ row_ce,
                        float* __restrict__ row_cnt,
                        int C) {
  __shared__ __align__(16) float stage[WAVES][2][WCHUNK];  // 64 KB double buffer
  __shared__ float red_m[WAVES], red_s[WAVES];
  __shared__ int   red_g[WAVES], red_e[WAVES];
  __shared__ float s_xt;

  const int row  = blockIdx.x;
  const int tid  = threadIdx.x;
  const int w    = tid >> 5;
  const int lane = tid & 31;
  const float* rowp = logits + (size_t)row * (size_t)C;
  const long long t = target[row];
  const int tcol = (int)t;            // C < 2^31, safe

  if (tid == 0) s_xt = rowp[t];
  __syncthreads();
  const float xt = s_xt;

  float m = -FLT_MAX, s = 0.0f;
  int cgt = 0, ceq = 0;

  const int nfull  = C / CHUNK;
  const int base_w = w * WCHUNK + lane * 4;   // lane's offset within a block chunk

  if (nfull > 0) {
    // prologue: stage chunk 0 into buffer 0 (8 x b128 per lane)
    {
      const float* g0 = rowp + base_w;
#pragma unroll
      for (int q = 0; q < 8; ++q)
        async_ld16(g0 + q * 128, &stage[w][0][q * 128 + lane * 4]);
    }
    for (int st = 0; st < nfull; ++st) {
      const int  cur  = st & 1;
      const bool more = (st + 1) < nfull;
      if (more) {
        const float* gn = rowp + (size_t)(st + 1) * CHUNK + base_w;
#pragma unroll
        for (int q = 0; q < 8; ++q)
          async_ld16(gn + q * 128, &stage[w][cur ^ 1][q * 128 + lane * 4]);
        wait_async8();   // 16 outstanding; <=8 means current chunk landed in LDS
      } else {
        wait_async0();
      }

      // pull this lane's 32 staged values out of LDS
      float vv[LANEF];
#pragma unroll
      for (int q = 0; q < 8; ++q) {
        const float4 v = *reinterpret_cast<const float4*>(&stage[w][cur][q * 128 + lane * 4]);
        vv[q * 4 + 0] = v.x; vv[q * 4 + 1] = v.y;
        vv[q * 4 + 2] = v.z; vv[q * 4 + 3] = v.w;
      }

      // chunk-local max, single rescale of running sum
      float mc = vv[0];
#pragma unroll
      for (int i = 1; i < LANEF; ++i) mc = fmaxf(mc, vv[i]);
      const float newM = fmaxf(m, mc);
      s *= __expf(m - newM);
      m = newM;

      // exp-sum with 2-way ILP
      float s0 = 0.0f, s1 = 0.0f;
#pragma unroll
      for (int i = 0; i < LANEF; i += 2) {
        s0 += __expf(vv[i]     - m);
        s1 += __expf(vv[i + 1] - m);
      }
      s += s0 + s1;

      // rank counters (32-bit columns)
      const int stepbase = st * CHUNK + base_w;   // col = stepbase + q*128 + i
#pragma unroll
      for (int q = 0; q < 8; ++q) {
#pragma unroll
        for (int i = 0; i < 4; ++i) {
          const float x = vv[q * 4 + i];
          const int col = stepbase + q * 128 + i;
          cgt += (x > xt) ? 1 : 0;
          ceq += ((x == xt) && (col < tcol)) ? 1 : 0;
        }
      }
    }
  }

  // tail (guarded direct loads; avoids async OOB past the last row)
  for (int j = nfull * CHUNK + tid; j < C; j += THREADS) {
    const float x = rowp[j];
    if (x > m) { s = s * __expf(m - x) + 1.0f; m = x; }
    else       { s += __expf(x - m); }
    cgt += (x > xt) ? 1 : 0;
    ceq += ((x == xt) && (j < tcol)) ? 1 : 0;
  }

  // wave32 butterfly reduction
#pragma unroll
  for (int off = 16; off > 0; off >>= 1) {
    float m2 = __shfl_xor(m, off, 32);
    float s2 = __shfl_xor(s, off, 32);
    sm_merge(m, s, m2, s2);
    cgt += __shfl_xor(cgt, off, 32);
    ceq += __shfl_xor(ceq, off, 32);
  }
  if (lane == 0) { red_m[w] = m; red_s[w] = s; red_g[w] = cgt; red_e[w] = ceq; }
  __syncthreads();

  if (tid == 0) {
    float M = red_m[0], S = red_s[0];
    int G = red_g[0], E = red_e[0];
#pragma unroll
    for (int i = 1; i < WAVES; ++i) {
      sm_merge(M, S, red_m[i], red_s[i]);
      G += red_g[i]; E += red_e[i];
    }
    const float ce   = (M + __logf(S)) - xt;   // -log_softmax at target
    const int   rank = G + E;
    const int   k    = kptr[0];
    const bool  mis  = !(rank < k);
    row_ce[row]  = mis ? ce   : 0.0f;
    row_cnt[row] = mis ? 1.0f : 0.0f;
  }
}

__global__ __launch_bounds__(256)
void topk_ce_final_kernel(const float* __restrict__ row_ce,
                          const float* __restrict__ row_cnt,
                          float* __restrict__ out, int B) {
  __shared__ float sa[256], sb[256];
  float a = 0.0f, b = 0.0f;
  for (int i = threadIdx.x; i < B; i += 256) { a += row_ce[i]; b += row_cnt[i]; }
  sa[threadIdx.x] = a; sb[threadIdx.x] = b;
  __syncthreads();
  for (int off = 128; off > 0; off >>= 1) {
    if (threadIdx.x < off) {
      sa[threadIdx.x] += sa[threadIdx.x + off];
      sb[threadIdx.x] += sb[threadIdx.x + off];
    }
    __syncthreads();
  }
  if (threadIdx.x == 0) out[0] = (sb[0] > 0.0f) ? (sa[0] / sb[0]) : 0.0f;
}

extern "C" void kernel_launch(void* const* d_in, const int* in_sizes, int n_in,
                              void* d_out, int out_size, void* d_ws, size_t ws_size,
                              hipStream_t stream) {
  const float*     logits = (const float*)d_in[0];
  const long long* target = (const long long*)d_in[1];
  const int*       kptr   = (const int*)d_in[2];
  const int B = in_sizes[1];
  const int C = in_sizes[0] / B;

  float* row_ce  = (float*)d_ws;
  float* row_cnt = row_ce + B;

  topk_ce_row_kernel<<<B, THREADS, 0, stream>>>(logits, target, kptr, row_ce, row_cnt, C);
  topk_ce_final_kernel<<<1, 256, 0, stream>>>(row_ce, row_cnt, (float*)d_out, B);
}